// BayesFittingNet_13967233647509
// MI455X (gfx1250) — compile-verified
//
#include <hip/hip_runtime.h>
#include <math.h>

// MI455X (gfx1250) implementation.
// Roofline: obs = 2M x 16 f32 = 128 MB streamed once; ~0.8 FLOP/byte -> purely
// HBM-bound (~5.5us at 23.3 TB/s). Compute is free, so the per-tile quadratic
// form is done with V_WMMA_F32_16X16X4_F32 (exact f32, matches reference):
//   T = D_tile(16x16) @ Sigma_post(16x16)   via 4 accumulating K=4 WMMAs
//   sum of per-row quad forms over the tile = sum_elements(T .* D_tile)
// which needs no transpose and no per-row cross-lane reduction.

typedef __attribute__((ext_vector_type(2))) float v2f;
typedef __attribute__((ext_vector_type(8))) float v8f;

#define WS_S     0     // 256 floats: Sigma_post, row-major 16x16
#define WS_MU    256   // 16 floats : mu_post
#define WS_CONST 272   // 1 float   : N*16*log(2pi) + 0.5*N*logdet
#define WS_ACC   273   // 1 float   : quad-sum accumulator (zeroed each launch)

// ---------------------------------------------------------------------------
// Kernel 1: all the tiny 2x2-block linear algebra (single thread).
// ---------------------------------------------------------------------------
__global__ void bayes_setup(const float* __restrict__ mu_like,
                            const float* __restrict__ pose,
                            const float* __restrict__ sp_params,
                            const float* __restrict__ sl_params,
                            float* __restrict__ ws, int n_obs) {
  if (threadIdx.x != 0 || blockIdx.x != 0) return;
  float* S  = ws + WS_S;
  float* mp = ws + WS_MU;
  for (int i = 0; i < 256; ++i) S[i] = 0.f;
  ws[WS_ACC] = 0.f;   // deterministic accumulator init every launch

  // mu_prior = rotate mu_like points about centroid, translate to pose[:2]
  float px[8], py[8], cx = 0.f, cy = 0.f;
  for (int j = 0; j < 8; ++j) {
    px[j] = mu_like[2*j]; py[j] = mu_like[2*j+1];
    cx += px[j]; cy += py[j];
  }
  cx *= 0.125f; cy *= 0.125f;
  float ct = cosf(pose[2]), st = sinf(pose[2]);
  float mu_prior[16];
  for (int j = 0; j < 8; ++j) {
    float dx = px[j] - cx, dy = py[j] - cy;
    mu_prior[2*j]   = ct*dx - st*dy + pose[0];
    mu_prior[2*j+1] = st*dx + ct*dy + pose[1];
  }

  float logdet = 0.f;
  for (int b = 0; b < 8; ++b) {
    // prior block covariance  B@B^T + eps*I  and its inverse
    float p0 = sp_params[4*b], p1 = sp_params[4*b+1];
    float p2 = sp_params[4*b+2], p3 = sp_params[4*b+3];
    float a00 = p0*p0 + p1*p1 + 1e-6f;
    float a01 = p0*p2 + p1*p3;
    float a11 = p2*p2 + p3*p3 + 1e-6f;
    float ida = 1.f / (a00*a11 - a01*a01);
    float Pp00 = a11*ida, Pp01 = -a01*ida, Pp11 = a00*ida;
    // likelihood block covariance and inverse
    float q0 = sl_params[4*b], q1 = sl_params[4*b+1];
    float q2 = sl_params[4*b+2], q3 = sl_params[4*b+3];
    float b00 = q0*q0 + q1*q1 + 1e-6f;
    float b01 = q0*q2 + q1*q3;
    float b11 = q2*q2 + q3*q3 + 1e-6f;
    float idb = 1.f / (b00*b11 - b01*b01);
    float Pl00 = b11*idb, Pl01 = -b01*idb, Pl11 = b00*idb;
    // posterior precision and Sigma_post block
    float c00 = Pp00 + Pl00, c01 = Pp01 + Pl01, c11 = Pp11 + Pl11;
    float detc = c00*c11 - c01*c01;
    logdet += logf(detc);
    float idc = 1.f / detc;
    float s00 = c11*idc, s01 = -c01*idc, s11 = c00*idc;
    S[(2*b  )*16 + 2*b    ] = s00;
    S[(2*b  )*16 + 2*b + 1] = s01;
    S[(2*b+1)*16 + 2*b    ] = s01;
    S[(2*b+1)*16 + 2*b + 1] = s11;
    // mu_post block = Sigma_post @ (P_prior mu_prior + P_like mu_like)
    float u0 = mu_prior[2*b], u1 = mu_prior[2*b+1];
    float w0 = mu_like[2*b],  w1 = mu_like[2*b+1];
    float r0 = Pp00*u0 + Pp01*u1 + Pl00*w0 + Pl01*w1;
    float r1 = Pp01*u0 + Pp11*u1 + Pl01*w0 + Pl11*w1;
    mp[2*b]   = s00*r0 + s01*r1;
    mp[2*b+1] = s01*r0 + s11*r1;
  }
  const float LOG2PI = 1.8378770664093453f;
  ws[WS_CONST] = (float)n_obs * (16.f*LOG2PI + 0.5f*logdet);
}

// ---------------------------------------------------------------------------
// Kernel 2: HBM-bound streaming quad-form reduction using f32 WMMA.
// One wave = one 16x16 obs tile per iteration (1 KB unique HBM bytes).
// ---------------------------------------------------------------------------
__global__ void __launch_bounds__(256)
bayes_quad(const float* __restrict__ obs, const float* __restrict__ ws,
           float* __restrict__ acc_out, int tiles) {
  const int lane   = threadIdx.x & 31;
  const int wave   = (int)((blockIdx.x * blockDim.x + threadIdx.x) >> 5);
  const int nwaves = (int)((gridDim.x * blockDim.x) >> 5);
  const int hi = lane >> 4;      // lane half (selects M/K split per ISA layout)
  const int n  = lane & 15;      // column (B/C/D N index) or row (A M index)

  const float* S  = ws + WS_S;
  const float* mu = ws + WS_MU;

  // Preload Sigma_post in B-matrix layout (4x16 f32 chunks: VGPR.x = K even,
  // VGPR.y = K odd, K offset +2 for lanes 16-31) and mu_post views.
  v2f Bm[4], muA[4];
#pragma unroll
  for (int c = 0; c < 4; ++c) {
    int k = 4*c + 2*hi;
    Bm[c].x  = S[(k  )*16 + n];
    Bm[c].y  = S[(k+1)*16 + n];
    muA[c].x = mu[k];
    muA[c].y = mu[k+1];
  }
  const float mun = mu[n];

  float q = 0.f;
  for (int t = wave; t < tiles; t += nwaves) {  // wave-uniform: EXEC stays all-1s
    const float* base = obs + (size_t)t * 256;

    // D tile in A-matrix layout (16x4 f32 per chunk): lane = row, b64 loads.
    v2f A[4];
#pragma unroll
    for (int c = 0; c < 4; ++c) {
      int k = 4*c + 2*hi;
      v2f d = *(const v2f*)(base + n*16 + k);
      A[c].x = d.x - muA[c].x;
      A[c].y = d.y - muA[c].y;
    }
    // Same tile in C/D layout: VGPR r = row r+8*hi, lane%16 = column.
    float dC[8];
#pragma unroll
    for (int r = 0; r < 8; ++r)
      dC[r] = base[(r + 8*hi)*16 + n] - mun;

    // T = D @ Sigma_post  (four accumulating 16x16x4 f32 WMMAs)
    v8f acc = {0.f,0.f,0.f,0.f,0.f,0.f,0.f,0.f};
#pragma unroll
    for (int c = 0; c < 4; ++c)
      acc = __builtin_amdgcn_wmma_f32_16x16x4_f32(
          false, A[c], false, Bm[c], (short)0, acc, false, false);

    // grand-sum of T .* D == sum of per-row quadratic forms over the tile
#pragma unroll
    for (int r = 0; r < 8; ++r)
      q += acc[r] * dC[r];
  }

  // wave32 reduction, one atomic per wave
#pragma unroll
  for (int off = 16; off >= 1; off >>= 1)
    q += __shfl_xor(q, off, 32);
  if (lane == 0)
    atomicAdd(acc_out, q);
}

// ---------------------------------------------------------------------------
// Kernel 3: tail rows (N % 16) + final scalar assembly.
// ---------------------------------------------------------------------------
__global__ void bayes_finalize(const float* __restrict__ obs,
                               const float* __restrict__ ws,
                               float* __restrict__ out, int n_obs, int tiles) {
  if (threadIdx.x != 0 || blockIdx.x != 0) return;
  const float* S  = ws + WS_S;
  const float* mu = ws + WS_MU;
  float tail = 0.f;
  for (int i = tiles * 16; i < n_obs; ++i) {
    const float* row = obs + (size_t)i * 16;
    float qr = 0.f;
    for (int b = 0; b < 8; ++b) {
      float d0 = row[2*b]   - mu[2*b];
      float d1 = row[2*b+1] - mu[2*b+1];
      float s00 = S[(2*b)*16 + 2*b];
      float s01 = S[(2*b)*16 + 2*b + 1];
      float s11 = S[(2*b+1)*16 + 2*b + 1];
      qr += s00*d0*d0 + 2.f*s01*d0*d1 + s11*d1*d1;
    }
    tail += qr;
  }
  out[0] = ws[WS_CONST] + 0.5f * (ws[WS_ACC] + tail);
}

// ---------------------------------------------------------------------------
extern "C" void kernel_launch(void* const* d_in, const int* in_sizes, int n_in,
                              void* d_out, int out_size, void* d_ws, size_t ws_size,
                              hipStream_t stream) {
  const float* obs     = (const float*)d_in[0];  // (N,16) f32
  const float* mu_like = (const float*)d_in[1];  // (16,)
  const float* pose    = (const float*)d_in[2];  // (3,)
  const float* spp     = (const float*)d_in[3];  // (32,)
  const float* slp     = (const float*)d_in[4];  // (32,)
  float* ws  = (float*)d_ws;
  float* out = (float*)d_out;

  const int n_obs = in_sizes[0] / 16;
  const int tiles = n_obs / 16;

  bayes_setup<<<1, 1, 0, stream>>>(mu_like, pose, spp, slp, ws, n_obs);

  int blocks = 2048;                         // 16384 waves -> ~8 tiles each
  int maxb = (tiles + 7) / 8;                // never launch idle-only blocks
  if (maxb < 1) maxb = 1;
  if (blocks > maxb) blocks = maxb;
  bayes_quad<<<blocks, 256, 0, stream>>>(obs, ws, ws + WS_ACC, tiles);

  bayes_finalize<<<1, 1, 0, stream>>>(obs, ws, out, n_obs, tiles);
}